// Fully_connected_capsule_42786464203166
// MI455X (gfx1250) — compile-verified
//
#include <hip/hip_runtime.h>
#include <math.h>

// Capsule routing, fused 3-pass formulation for MI455X (gfx1250).
// Roofline: W = 514MB fp32 streamed once per routing iteration (x3) ~= 66us
// at 23.3 TB/s. u_hat / b_logits are never materialized: b_logits at iter k
// equals <sum of previous v's, u_hat>, so only vsum (1280 floats) persists.
// Per (o,i) the 16x8 @ 8x8 u_hat matmul maps to two chained
// V_WMMA_F32_16X16X4_F32 ops (M=16 v-rows, N=16 cols = 8 batches + 8 zero).
// K-order is permuted so WMMA#1 contracts d{0,1,4,5} and WMMA#2 d{2,3,6,7}:
// this makes each lane's two A(B) frags one contiguous 16B b128 load
// (full 512B (o,i) slice coalesced in a single instruction per wave).
// W loads are non-temporal: zero reuse, keeps x (25.7MB, reused all 3 passes)
// resident in the 192MB L2.

#define B_ 8
#define O_ 10
#define I_ 100352
#define V_ 16
#define D_ 8
#define WAVES 8
#define IPW 64                  // input capsules per wave
#define NBLK 196                // 196 blocks * 8 waves * 64 i = 100352
#define SOUT (O_ * V_ * B_)     // 1280 floats per partial

typedef __attribute__((ext_vector_type(2))) float v2f;
typedef __attribute__((ext_vector_type(4))) float v4f;
typedef __attribute__((ext_vector_type(8))) float v8f;

__global__ __launch_bounds__(256) void caps_zero(float* __restrict__ vsum) {
    int t = threadIdx.x;
    for (int j = t; j < SOUT; j += 256) vsum[j] = 0.f;
}

__global__ __launch_bounds__(256) void caps_pass(
    const float* __restrict__ x, const float* __restrict__ W,
    const float* __restrict__ vsum, float* __restrict__ gpart)
{
    // vsum staged in LDS as [b(16, padded with zeros)][o][v]
    __shared__ float vs[16 * O_ * V_];          // 10 KB
    __shared__ float sred[WAVES][SOUT];         // 40 KB, per-wave partials

    int t = threadIdx.x;
    for (int j = t; j < 16 * O_ * V_; j += 256)
        vs[j] = (j < SOUT) ? vsum[j] : 0.f;
    __syncthreads();

    const int w    = t >> 5;       // wave id in block (wave32)
    const int lane = t & 31;
    const int n    = lane & 15;    // WMMA column index (= batch b for n<8)
    const int hi   = lane >> 4;    // half-wave selector
    const int gw   = blockIdx.x * WAVES + w;
    const int i0   = gw * IPW;

    v8f sacc[O_];
    v8f zero8 = {};
#pragma unroll
    for (int o = 0; o < O_; ++o) sacc[o] = zero8;

    // Per-lane 16B offset inside one (o,i) 16x8 slice: row = lane&15 (M = v),
    // 4 contiguous d's starting at 4*hi. (K-permuted WMMA contraction.)
    const size_t aoff = (size_t)n * D_ + (size_t)hi * 4;

    for (int ii = 0; ii < IPW; ++ii) {
        const int i = i0 + ii;

        // B frags: B[k,n] = x[b=n, i, d]; columns n>=8 are zero.
        v2f b0 = {}, b1 = {};
        if (n < B_) {
            const v4f xq = *(const v4f*)(x + ((size_t)n * I_ + i) * D_ + hi * 4);
            b0 = __builtin_shufflevector(xq, xq, 0, 1);   // d = 4hi, 4hi+1
            b1 = __builtin_shufflevector(xq, xq, 2, 3);   // d = 4hi+2, 4hi+3
        }

        v2f A0[O_], A1[O_];
        float blog[O_];
#pragma unroll
        for (int o = 0; o < O_; ++o) {
            // one coalesced, non-temporal b128 per (o,i) per lane
            const v4f wv = __builtin_nontemporal_load(
                (const v4f*)(W + ((size_t)o * I_ + i) * (V_ * D_) + aoff));
            A0[o] = __builtin_shufflevector(wv, wv, 0, 1);
            A1[o] = __builtin_shufflevector(wv, wv, 2, 3);

            // u_hat tile: C[m=v, n=b] = sum_d W[o,i,v,d] * x[b,i,d]
            v8f c = {};
            c = __builtin_amdgcn_wmma_f32_16x16x4_f32(false, A0[o], false, b0,
                                                      (short)0, c, false, false);
            c = __builtin_amdgcn_wmma_f32_16x16x4_f32(false, A1[o], false, b1,
                                                      (short)0, c, false, false);

            // b_logit partial = <vsum[b,o, v-range-of-this-half], C>
            const v4f* vp = (const v4f*)&vs[(n * O_ + o) * V_ + hi * 8];
            v4f va = vp[0], vb = vp[1];
            blog[o] = va[0]*c[0] + va[1]*c[1] + va[2]*c[2] + va[3]*c[3]
                    + vb[0]*c[4] + vb[1]*c[5] + vb[2]*c[6] + vb[3]*c[7];
        }

        // join the two half-wave v-ranges: full <vsum, u_hat> per (b,o)
#pragma unroll
        for (int o = 0; o < O_; ++o)
            blog[o] += __shfl_xor(blog[o], 16, 32);

        // softmax over the 10 output capsules (local per (b,i) lane)
        float m = blog[0];
#pragma unroll
        for (int o = 1; o < O_; ++o) m = fmaxf(m, blog[o]);
        float ssum = 0.f;
        float cw[O_];
#pragma unroll
        for (int o = 0; o < O_; ++o) { cw[o] = __expf(blog[o] - m); ssum += cw[o]; }
        const float inv = 1.f / ssum;

        // s[b,o,v] += c[b,o,i] * u_hat : scale B columns by c, accumulate in C
#pragma unroll
        for (int o = 0; o < O_; ++o) {
            const float co = cw[o] * inv;
            v2f sb0 = b0 * co;
            v2f sb1 = b1 * co;
            sacc[o] = __builtin_amdgcn_wmma_f32_16x16x4_f32(false, A0[o], false, sb0,
                                                            (short)0, sacc[o], false, false);
            sacc[o] = __builtin_amdgcn_wmma_f32_16x16x4_f32(false, A1[o], false, sb1,
                                                            (short)0, sacc[o], false, false);
        }
    }

    // deterministic block reduction: each wave dumps its s partial to LDS
    if (n < B_) {
#pragma unroll
        for (int o = 0; o < O_; ++o)
#pragma unroll
            for (int r = 0; r < 8; ++r)
                sred[w][(o * V_ + r + 8 * hi) * B_ + n] = sacc[o][r];
    }
    __syncthreads();

    float* gp = gpart + (size_t)blockIdx.x * SOUT;
    for (int j = t; j < SOUT; j += 256) {
        float a = 0.f;
#pragma unroll
        for (int ww = 0; ww < WAVES; ++ww) a += sred[ww][j];
        gp[j] = a;
    }
}

__global__ __launch_bounds__(256) void caps_reduce(
    const float* __restrict__ gpart, float* __restrict__ vsum,
    float* __restrict__ out, int final_iter)
{
    __shared__ float sm[SOUT];   // s[b,o,v] stored as [o][v][b]
    int t = threadIdx.x;
    for (int j = t; j < SOUT; j += 256) {
        float a = 0.f;
        for (int g = 0; g < NBLK; ++g)      // fixed order -> deterministic
            a += gpart[(size_t)g * SOUT + j];
        sm[j] = a;
    }
    __syncthreads();

    if (t < B_ * O_) {
        const int o = t / B_, b = t % B_;
        float sq = 0.f;
#pragma unroll
        for (int v = 0; v < V_; ++v) {
            float s = sm[(o * V_ + v) * B_ + b];
            sq += s * s;
        }
        const float norm = sqrtf(sq);
        const float f = (sq / (1.f + sq)) / (norm + 1e-7f);
        if (final_iter) {
#pragma unroll
            for (int v = 0; v < V_; ++v)
                out[(b * O_ + o) * V_ + v] = f * sm[(o * V_ + v) * B_ + b];
        } else {
#pragma unroll
            for (int v = 0; v < V_; ++v)
                vsum[(b * O_ + o) * V_ + v] += f * sm[(o * V_ + v) * B_ + b];
        }
    }
}

extern "C" void kernel_launch(void* const* d_in, const int* in_sizes, int n_in,
                              void* d_out, int out_size, void* d_ws, size_t ws_size,
                              hipStream_t stream) {
    (void)in_sizes; (void)n_in; (void)out_size; (void)ws_size;
    const float* x = (const float*)d_in[0];
    const float* W = (const float*)d_in[1];
    float* out   = (float*)d_out;
    float* vsum  = (float*)d_ws;            // 1280 floats
    float* gpart = (float*)d_ws + 2048;     // 196 * 1280 floats (~1 MB)

    caps_zero<<<1, 256, 0, stream>>>(vsum);
    for (int r = 0; r < 3; ++r) {
        caps_pass<<<NBLK, 256, 0, stream>>>(x, W, vsum, gpart);
        caps_reduce<<<1, 256, 0, stream>>>(gpart, vsum, out, (r == 2) ? 1 : 0);
    }
}